// SECOND_BEAM_SEARCH_8091718386201
// MI455X (gfx1250) — compile-verified
//
#include <hip/hip_runtime.h>
#include <hip/hip_bf16.h>
#include <math.h>

// ---------------- problem constants ----------------
#define NUM_KV   48
#define BEAM     10
#define HEADS    16
#define SEQ      448
#define HDIM     64
#define VOCAB    50257
#define DEC_LEN  447

#define SLICE_F   (HEADS * SEQ * HDIM)   // 458752 floats per (layer,beam) slice
#define SLICE_V4  (SLICE_F / 4)          // 114688 float4
#define PARTS     7                      // blocks per slice
#define PART_V4   (SLICE_V4 / PARTS)     // 16384 float4 per block
#define COPY_T    256                    // threads per copy block
#define CHUNK     4                      // float4 per thread per phase
#define PHASE_V4  (COPY_T * CHUNK)       // 1024 float4 = 16KB per LDS buffer
#define PHASES    (PART_V4 / PHASE_V4)   // 16

// vocab partitioning for the top-k kernel
#define NB        8                            // blocks per beam
#define CHUNK_A   ((VOCAB + NB - 1) / NB)      // 6283 elements per block

// d_out float offsets (outputs concatenated flat in return order)
#define OFF_KV      0LL
#define OFF_TOK     220200960LL                  // 48*10*16*448*64
#define OFF_SAVE    (OFF_TOK + 10LL)
#define OFF_RP      (OFF_SAVE + 10LL * 448LL)
#define OFF_PROB    (OFF_RP + 10LL * VOCAB)
#define OFF_MAXIDX  (OFF_PROB + 10LL)

// ------------------------------------------------------------------
// Kernel A: per-(beam, vocab-chunk) partials. grid = 10*NB, block=256.
// Stages the chunk of x = logits*rp in LDS, computes (max, sumexp)
// partials and the chunk's top-10 via LDS argmax rounds. No register
// lists -> no spills; one global read of the data.
// ------------------------------------------------------------------
__global__ void beam_partial_kernel(const float* __restrict__ logits,
                                    const float* __restrict__ rp,
                                    float* __restrict__ blk_m,
                                    float* __restrict__ blk_s,
                                    float* __restrict__ blk_val,
                                    int*   __restrict__ blk_idx) {
    const int T    = 256;
    const int beam = blockIdx.x / NB;
    const int blk  = blockIdx.x % NB;
    const int start = blk * CHUNK_A;
    const int count = min(CHUNK_A, VOCAB - start);
    const int tid  = threadIdx.x;
    const float* xl = logits + (size_t)beam * VOCAB + start;
    const float* xr = rp     + (size_t)beam * VOCAB + start;

    __shared__ float xbuf[CHUNK_A];   // ~25 KB
    __shared__ float red[T];
    __shared__ int   redi[T];

    // load chunk once, track local max
    float m = -INFINITY;
    for (int i = tid; i < count; i += T) {
        const float x = xl[i] * xr[i];
        xbuf[i] = x;
        m = fmaxf(m, x);
    }
    red[tid] = m; __syncthreads();
    for (int st = T / 2; st > 0; st >>= 1) {
        if (tid < st) red[tid] = fmaxf(red[tid], red[tid + st]);
        __syncthreads();
    }
    m = red[0]; __syncthreads();

    // sumexp over LDS copy
    float s = 0.f;
    for (int i = tid; i < count; i += T) s += expf(xbuf[i] - m);
    red[tid] = s; __syncthreads();
    for (int st = T / 2; st > 0; st >>= 1) {
        if (tid < st) red[tid] += red[tid + st];
        __syncthreads();
    }
    if (tid == 0) { blk_m[blockIdx.x] = m; blk_s[blockIdx.x] = red[0]; }
    __syncthreads();

    // 10 argmax rounds over the LDS chunk (lower index wins ties)
    for (int r = 0; r < 10; ++r) {
        float bv = -INFINITY; int bp = 0x7fffffff;
        for (int i = tid; i < count; i += T) {
            const float v = xbuf[i];
            if (v > bv || (v == bv && i < bp)) { bv = v; bp = i; }
        }
        red[tid] = bv; redi[tid] = bp; __syncthreads();
        for (int st = T / 2; st > 0; st >>= 1) {
            if (tid < st) {
                const float ov = red[tid + st]; const int op = redi[tid + st];
                if (ov > red[tid] || (ov == red[tid] && op < redi[tid])) {
                    red[tid] = ov; redi[tid] = op;
                }
            }
            __syncthreads();
        }
        if (tid == 0) {
            blk_val[blockIdx.x * 10 + r] = red[0];
            blk_idx[blockIdx.x * 10 + r] = start + redi[0];
            xbuf[redi[0]] = -INFINITY;        // remove winner
        }
        __syncthreads();
    }
}

// ------------------------------------------------------------------
// Kernel A2: combine partials -> per-beam logZ + top-10, then the
// global 100 -> 10 beam selection. Serial (trivial work).
// ------------------------------------------------------------------
__global__ void combine_select_kernel(const float* __restrict__ blk_m,
                                      const float* __restrict__ blk_s,
                                      const float* __restrict__ blk_val,
                                      const int*   __restrict__ blk_idx,
                                      const float* __restrict__ prev,
                                      int* __restrict__ beam_index,
                                      int* __restrict__ tokens,
                                      float* __restrict__ out) {
    if (threadIdx.x != 0) return;
    float cand_prob[100]; int cand_tok[100];

#pragma unroll 1
    for (int b = 0; b < BEAM; ++b) {
        // exact logsumexp from block partials
        float M = -INFINITY;
#pragma unroll 1
        for (int k = 0; k < NB; ++k) M = fmaxf(M, blk_m[b * NB + k]);
        float S = 0.f;
#pragma unroll 1
        for (int k = 0; k < NB; ++k) S += blk_s[b * NB + k] * expf(blk_m[b * NB + k] - M);
        const float logZ = M + logf(S);
        const float pp   = prev[b];

        // top-10 of this beam's 80 block-candidates
        float v[NB * 10]; int ix[NB * 10];
#pragma unroll 1
        for (int k = 0; k < NB * 10; ++k) {
            v[k]  = blk_val[b * NB * 10 + k];
            ix[k] = blk_idx[b * NB * 10 + k];
        }
#pragma unroll 1
        for (int r = 0; r < 10; ++r) {
            float bv = -INFINITY; int bi = 0x7fffffff; int bp = 0;
#pragma unroll 1
            for (int k = 0; k < NB * 10; ++k) {
                if (v[k] > bv || (v[k] == bv && ix[k] < bi)) { bv = v[k]; bi = ix[k]; bp = k; }
            }
            v[bp] = -INFINITY; ix[bp] = 0x7fffffff;
            cand_prob[b * 10 + r] = bv - logZ + pp;
            cand_tok[b * 10 + r]  = bi;
        }
    }

    // final selection: 100 candidates -> 10 beams
    bool used[100];
#pragma unroll 1
    for (int i = 0; i < 100; ++i) used[i] = false;
#pragma unroll 1
    for (int j = 0; j < 10; ++j) {
        float bv = -INFINITY; int bf = 0;
#pragma unroll 1
        for (int f = 0; f < 100; ++f) {
            if (used[f]) continue;
            const float vv = cand_prob[f];
            if (vv > bv) { bv = vv; bf = f; }   // strict > keeps lowest flat index on ties
        }
        used[bf] = true;
        const int bi = bf / 10;
        const int tk = cand_tok[bf];
        beam_index[j] = bi;
        tokens[j]     = tk;
        out[OFF_TOK  + j] = (float)tk;
        out[OFF_PROB + j] = bv;
        if (j == 0) out[OFF_MAXIDX] = (float)tk;
    }
}

// ------------------------------------------------------------------
// Kernel B: kv gather-copy via gfx1250 async global<->LDS DMA path,
// double-buffered: loads of phase p+1 overlap stores of phase p,
// one s_wait_asynccnt per phase. Each wave owns disjoint LDS slots,
// so no block barriers are needed.
// ------------------------------------------------------------------
__device__ __forceinline__ void async_tile_load(unsigned lds, unsigned off,
                                                unsigned long long src) {
#pragma unroll
    for (int k = 0; k < CHUNK; ++k) {
        asm volatile("global_load_async_to_lds_b128 %0, %1, %2"
                     :: "v"(lds + (unsigned)(k * COPY_T) * 16u),
                        "v"(off + (unsigned)(k * COPY_T) * 16u),
                        "s"(src)
                     : "memory");
    }
}

__device__ __forceinline__ void async_tile_store(unsigned lds, unsigned off,
                                                 unsigned long long dst) {
#pragma unroll
    for (int k = 0; k < CHUNK; ++k) {
        asm volatile("global_store_async_from_lds_b128 %0, %1, %2"
                     :: "v"(off + (unsigned)(k * COPY_T) * 16u),
                        "v"(lds + (unsigned)(k * COPY_T) * 16u),
                        "s"(dst)
                     : "memory");
    }
}

__global__ void kv_gather_async_kernel(const float* __restrict__ kv,
                                       float* __restrict__ out,
                                       const int* __restrict__ beam_index) {
    __shared__ float4 stage[2][PHASE_V4];        // 2 x 16KB
    const unsigned bid   = blockIdx.x;
    const unsigned slice = bid / PARTS;          // 0..479  (= l*10 + b)
    const unsigned part  = bid % PARTS;          // 0..6
    const unsigned l     = slice / 10u;
    const unsigned b     = slice - l * 10u;
    const int sb = beam_index[b];

    const unsigned long long src =
        (unsigned long long)(kv  + ((size_t)(l * 10u + (unsigned)sb)) * SLICE_F);
    const unsigned long long dst =
        (unsigned long long)(out + ((size_t)slice) * SLICE_F);

    // low 32 bits of a generic LDS pointer == LDS byte offset on gfx1250
    const unsigned lds0 = (unsigned)(size_t)(&stage[0][threadIdx.x]);
    const unsigned lds1 = (unsigned)(size_t)(&stage[1][threadIdx.x]);
    const unsigned base = (part * PART_V4 + threadIdx.x) * 16u;   // byte offset

    // prologue: fill buffer 0 with phase 0
    async_tile_load(lds0, base, src);

    for (int p = 0; p < PHASES; ++p) {
        const unsigned off = base + (unsigned)(p * PHASE_V4) * 16u;
        // all prior stores (S_{p-1}) and current loads (L_p) complete here
        asm volatile("s_wait_asynccnt 0" ::: "memory");
        const unsigned cur = (p & 1) ? lds1 : lds0;
        async_tile_store(cur, off, dst);
        if (p + 1 < PHASES) {
            const unsigned nxt = (p & 1) ? lds0 : lds1;
            async_tile_load(nxt, off + (unsigned)PHASE_V4 * 16u, src);
        }
    }
    asm volatile("s_wait_asynccnt 0" ::: "memory");   // drain final stores
}

// ------------------------------------------------------------------
// Kernel C: repeat_penality gather + penalty at the selected token
// ------------------------------------------------------------------
__global__ void rp_gather_kernel(const float* __restrict__ rp_in,
                                 const int* __restrict__ beam_index,
                                 const int* __restrict__ tokens,
                                 const float* __restrict__ pv,
                                 float* __restrict__ out_rp) {
    const int b = blockIdx.y;
    const int v = blockIdx.x * blockDim.x + threadIdx.x;
    if (v >= VOCAB) return;
    const int sb = beam_index[b];
    float val = rp_in[(size_t)sb * VOCAB + v];
    if (v == tokens[b]) val *= pv[0];
    out_rp[(size_t)b * VOCAB + v] = val;
}

// ------------------------------------------------------------------
// Kernel D: save_id gather + append new token (as float values)
// ------------------------------------------------------------------
__global__ void save_id_gather_kernel(const int* __restrict__ save_id,
                                      const int* __restrict__ beam_index,
                                      const int* __restrict__ tokens,
                                      float* __restrict__ out_save) {
    const int b = blockIdx.x;
    const int t = threadIdx.x;       // 0..447
    const int sb = beam_index[b];
    const float v = (t < DEC_LEN) ? (float)save_id[(size_t)sb * DEC_LEN + t]
                                  : (float)tokens[b];
    out_save[(size_t)b * (DEC_LEN + 1) + t] = v;
}

// ------------------------------------------------------------------
extern "C" void kernel_launch(void* const* d_in, const int* in_sizes, int n_in,
                              void* d_out, int out_size, void* d_ws, size_t ws_size,
                              hipStream_t stream) {
    const float* kv      = (const float*)d_in[0];
    const float* logits  = (const float*)d_in[1];
    const int*   save_id = (const int*)  d_in[2];
    const float* rp      = (const float*)d_in[3];
    const float* prev    = (const float*)d_in[4];
    // d_in[5] = batch_indices (arange, unused)
    const float* pv      = (const float*)d_in[6];
    float* out = (float*)d_out;

    char* ws = (char*)d_ws;
    float* blk_m      = (float*)(ws + 0);      // 80 floats
    float* blk_s      = (float*)(ws + 512);    // 80 floats
    float* blk_val    = (float*)(ws + 1024);   // 800 floats
    int*   blk_idx    = (int*)  (ws + 4352);   // 800 ints
    int*   beam_index = (int*)  (ws + 7680);   // 10 ints
    int*   tokens     = (int*)  (ws + 7808);   // 10 ints

    beam_partial_kernel<<<BEAM * NB, 256, 0, stream>>>(logits, rp, blk_m, blk_s,
                                                       blk_val, blk_idx);
    combine_select_kernel<<<1, 32, 0, stream>>>(blk_m, blk_s, blk_val, blk_idx,
                                                prev, beam_index, tokens, out);
    kv_gather_async_kernel<<<NUM_KV * BEAM * PARTS, COPY_T, 0, stream>>>(
        kv, out + OFF_KV, beam_index);
    dim3 gc((VOCAB + 255) / 256, BEAM);
    rp_gather_kernel<<<gc, 256, 0, stream>>>(rp, beam_index, tokens, pv,
                                             out + OFF_RP);
    save_id_gather_kernel<<<BEAM, DEC_LEN + 1, 0, stream>>>(save_id, beam_index,
                                                            tokens, out + OFF_SAVE);
}